// Edge2NodeFeatures_31404800868643
// MI455X (gfx1250) — compile-verified
//
#include <hip/hip_runtime.h>

// Edge2NodeFeatures for MI455X (gfx1250, wave32).
// h = LayerNorm(ReLU(edge_attr @ W + b)); out = (scatter_add(h,dst)+scatter_add(h,src)) / clip(deg,1)
//
// Fully fused: fp32 WMMA GEMM (v_wmma_f32_16x16x4_f32) + in-register bias/ReLU/LayerNorm
// + dual-endpoint f32 atomic scatter (resolves in 192MB L2). Finalize kernel divides by degree.

#define NODE_DIM 128
#define EDGE_DIM 64
#define LDW 132                 // padded LDS row stride for W (bank-conflict-free B reads)
#define EDGES_PER_BLOCK 128     // 8 waves x 16 edges
#define LN_EPS 1e-5f

typedef __attribute__((ext_vector_type(2))) float v2f;
typedef __attribute__((ext_vector_type(8))) float v8f;

__global__ __launch_bounds__(256) void edge2node_main(
    const float* __restrict__ edge_attr,   // [E, 64]
    const float* __restrict__ W,           // [64, 128]
    const float* __restrict__ bvec,        // [128]
    const float* __restrict__ gamma,       // [128]
    const float* __restrict__ beta,        // [128]
    const int*   __restrict__ edge_index,  // [2, E] (src row, dst row)
    float*       __restrict__ out,         // [N, 128] (pre-zeroed)
    float*       __restrict__ degree,      // [N]      (pre-zeroed)
    int n_edges)
{
    __shared__ float ldsW[EDGE_DIM * LDW];

    const int tid = threadIdx.x;
    // Cooperative load of W (64x128) into padded LDS: fully coalesced.
    for (int i = tid; i < EDGE_DIM * NODE_DIM; i += 256) {
        int r = i >> 7;        // row (K)
        int c = i & 127;       // col (N)
        ldsW[r * LDW + c] = W[i];
    }
    __syncthreads();

    const int wave = tid >> 5;           // 0..7
    const int lane = tid & 31;
    const int m    = lane & 15;          // 0..15
    const int half = lane >> 4;          // 0 or 1
    const int base = blockIdx.x * EDGES_PER_BLOCK + wave * 16;
    if (base >= n_edges) return;         // wave-uniform (never taken for E=1.6M)

    // Per-lane per-N-tile parameters: column n = t*16 + m
    float bv[8], gv[8], bev[8];
#pragma unroll
    for (int t = 0; t < 8; ++t) {
        int n = t * 16 + m;
        bv[t]  = bvec[n];
        gv[t]  = gamma[n];
        bev[t] = beta[n];
    }

    // A fragments: 32-bit A-matrix 16x4 layout. Lane holds row M=m; VGPR pair holds
    // K = k*4 + 2*half and +1 (half-wave selects K offset 0 or 2).
    v2f a[16];
    const float* __restrict__ arow = edge_attr + (size_t)(base + m) * EDGE_DIM;
#pragma unroll
    for (int k = 0; k < 16; ++k) {
        int col = k * 4 + 2 * half;
        a[k].x = arow[col];
        a[k].y = arow[col + 1];
    }

    // GEMM: 8 N-tiles of 16x16, K=64 in 16 steps of 4.
    v8f acc[8];
#pragma unroll
    for (int t = 0; t < 8; ++t) {
        v8f cacc = {};
#pragma unroll
        for (int k = 0; k < 16; ++k) {
            int krow = k * 4 + 2 * half;   // B rows K=krow, krow+1 for this half-wave
            v2f bf;
            bf.x = ldsW[krow * LDW + t * 16 + m];
            bf.y = ldsW[(krow + 1) * LDW + t * 16 + m];
            cacc = __builtin_amdgcn_wmma_f32_16x16x4_f32(
                false, a[k], false, bf, (short)0, cacc, false, false);
        }
        acc[t] = cacc;
    }

    // C layout: acc[t][r] holds element (row M = r + 8*half, col N = t*16 + m).
    // Bias + ReLU + per-row LayerNorm stats.
    float s1[8], s2[8];
#pragma unroll
    for (int r = 0; r < 8; ++r) { s1[r] = 0.0f; s2[r] = 0.0f; }
#pragma unroll
    for (int t = 0; t < 8; ++t) {
#pragma unroll
        for (int r = 0; r < 8; ++r) {
            float x = acc[t][r] + bv[t];
            x = fmaxf(x, 0.0f);
            acc[t][r] = x;
            s1[r] += x;
            s2[r] += x * x;
        }
    }
    // Reduce across the 16 lanes of each half-wave (each half owns distinct rows).
#pragma unroll
    for (int r = 0; r < 8; ++r) {
#pragma unroll
        for (int off = 8; off >= 1; off >>= 1) {
            s1[r] += __shfl_xor(s1[r], off, 16);
            s2[r] += __shfl_xor(s2[r], off, 16);
        }
    }
    float mu[8], rs[8];
#pragma unroll
    for (int r = 0; r < 8; ++r) {
        float m1 = s1[r] * (1.0f / NODE_DIM);
        float var = s2[r] * (1.0f / NODE_DIM) - m1 * m1;
        mu[r] = m1;
        rs[r] = rsqrtf(var + LN_EPS);
    }
#pragma unroll
    for (int t = 0; t < 8; ++t) {
#pragma unroll
        for (int r = 0; r < 8; ++r) {
            acc[t][r] = (acc[t][r] - mu[r]) * rs[r] * gv[t] + bev[t];
        }
    }

    // Dual-endpoint scatter-add (fp32 atomics, hit in L2) + degree counting.
    const int* __restrict__ srcIdx = edge_index;            // edge_index[0]
    const int* __restrict__ dstIdx = edge_index + n_edges;  // edge_index[1]
#pragma unroll
    for (int r = 0; r < 8; ++r) {
        int e = base + r + 8 * half;
        int s = srcIdx[e];
        int d = dstIdx[e];
        float* orow_d = out + (size_t)d * NODE_DIM;
        float* orow_s = out + (size_t)s * NODE_DIM;
#pragma unroll
        for (int t = 0; t < 8; ++t) {
            float val = acc[t][r];
            int n = t * 16 + m;
            unsafeAtomicAdd(orow_d + n, val);
            unsafeAtomicAdd(orow_s + n, val);
        }
        if (m == 0) {
            unsafeAtomicAdd(&degree[s], 1.0f);
            unsafeAtomicAdd(&degree[d], 1.0f);
        }
    }
}

__global__ __launch_bounds__(256) void edge2node_finalize(
    float* __restrict__ out, const float* __restrict__ degree, int total)
{
    int i = blockIdx.x * blockDim.x + threadIdx.x;
    if (i < total) {
        float deg = fmaxf(degree[i >> 7], 1.0f);   // NODE_DIM == 128
        out[i] = out[i] / deg;
    }
}

extern "C" void kernel_launch(void* const* d_in, const int* in_sizes, int n_in,
                              void* d_out, int out_size, void* d_ws, size_t ws_size,
                              hipStream_t stream) {
    const float* edge_attr  = (const float*)d_in[0];
    const float* W          = (const float*)d_in[1];
    const float* bvec       = (const float*)d_in[2];
    const float* gamma      = (const float*)d_in[3];
    const float* beta       = (const float*)d_in[4];
    const int*   edge_index = (const int*)d_in[5];   // integer input -> const int*
    // d_in[6] = num_nodes scalar (also derivable from out_size)

    float* out    = (float*)d_out;
    float* degree = (float*)d_ws;

    const int n_edges   = in_sizes[5] / 2;
    const int num_nodes = out_size / NODE_DIM;

    // Graph-capture-legal async zero-fills (deterministic per call).
    hipMemsetAsync(d_out, 0, (size_t)out_size * sizeof(float), stream);
    hipMemsetAsync(d_ws, 0, (size_t)num_nodes * sizeof(float), stream);

    const int nblocks = (n_edges + EDGES_PER_BLOCK - 1) / EDGES_PER_BLOCK;  // 12500
    edge2node_main<<<nblocks, 256, 0, stream>>>(edge_attr, W, bvec, gamma, beta,
                                                edge_index, out, degree, n_edges);

    const int total = out_size;
    edge2node_finalize<<<(total + 255) / 256, 256, 0, stream>>>(out, degree, total);
}